// BitflipLayer_83932250898655
// MI455X (gfx1250) — compile-verified
//
#include <hip/hip_runtime.h>
#include <stdint.h>

// 128-bit vector type that lowers to b128 loads/stores and works with
// __builtin_nontemporal_store (HIP's uint4 is a struct, not a clang vector).
typedef uint32_t v4u __attribute__((ext_vector_type(4)));

// Per-element flip mask, statistically equivalent to 32 independent
// Bernoulli(0.001) bit flips:
//   K = popcount(mask) ~ Binomial(32, 0.001):
//     P(K>=1) = 1-(0.999)^32      = 0.0315090  -> 132139 / 2^22
//     P(K>=2) = 1-q^32-32*p*q^31  = 0.0004862  ->   2039 / 2^22
//   P(K>=3) ~ 5e-6 is dropped. Top 22 hash bits pick K, low 10 bits pick
//   up to two flip positions. ~12 VALU ops/element, fully branchless.
__device__ __forceinline__ uint32_t flip_mask(uint32_t eidx) {
  uint32_t x = eidx * 0x9E3779B9u;
  x ^= x >> 16; x *= 0x7feb352du;
  x ^= x >> 15; x *= 0x846ca68bu;
  x ^= x >> 16;
  const uint32_t hi = x >> 10;                 // 22 uniform bits
  const uint32_t b0 = 1u << (x & 31u);         // first flip position
  const uint32_t b1 = 1u << ((x >> 5) & 31u);  // second flip position
  uint32_t m = (hi < 132139u) ? b0 : 0u;       // K >= 1
  m = (hi < 2039u) ? (m | b1) : m;             // K >= 2
  return m;
}

// Streaming bit-flip: global -> (async DMA) -> LDS -> VGPR -> NT store.
// Double-buffered 2 x 256 x 16B LDS stage; each lane consumes only the 16B
// it staged itself, so synchronization is just s_wait_asynccnt (no barrier).
__global__ __launch_bounds__(256) void bitflip_async_kernel(
    const v4u* __restrict__ in, v4u* __restrict__ out, uint32_t n4) {
  __shared__ v4u stage[2][256];

  const uint32_t tid    = threadIdx.x;
  const uint32_t stride = gridDim.x * 256u;
  const uint32_t i0     = blockIdx.x * 256u + tid;
  const uint32_t last   = n4 - 1u;  // n4 >= 1 guaranteed by launcher

  // LDS byte address of this lane's slot in buffer 0 (addrspacecast of an
  // LDS pointer to flat keeps the LDS byte offset in the low 32 bits).
  const uint32_t lbase = (uint32_t)(uintptr_t)(&stage[0][tid]);

  // Prologue: stage the first tile (clamped) into buffer 0.
  {
    const uint32_t cur = (i0 <= last) ? i0 : last;
    const uint64_t ga  = (uint64_t)(uintptr_t)(in + cur);
    asm volatile("global_load_async_to_lds_b128 %0, %1, off"
                 :: "v"(lbase), "v"(ga) : "memory");
  }

  uint32_t buf = 0u;
  for (uint32_t i = i0; i < n4; i += stride) {
    // Kick off the next tile into the other buffer. Clamp the address so we
    // never read out of bounds; clamped lanes' data is never consumed.
    uint32_t nxt = i + stride;
    nxt = (nxt <= last) ? nxt : last;
    const uint64_t ga = (uint64_t)(uintptr_t)(in + nxt);
    const uint32_t lnext = lbase + (buf ^ 1u) * 4096u;
    asm volatile("global_load_async_to_lds_b128 %0, %1, off"
                 :: "v"(lnext), "v"(ga) : "memory");

    // Wait until this iteration's tile (the older of the two outstanding
    // async loads) has landed in LDS; the next tile stays in flight.
    asm volatile("s_wait_asynccnt 0x1" ::: "memory");

    v4u v = stage[buf][tid];  // ds_load_b128, own lane's data only

    const uint32_t e = i * 4u;
    v.x ^= flip_mask(e);
    v.y ^= flip_mask(e + 1u);
    v.z ^= flip_mask(e + 2u);
    v.w ^= flip_mask(e + 3u);

    // Streaming store: 256 MiB working set exceeds the 192 MB L2 -> NT hint.
    __builtin_nontemporal_store(v, out + i);

    buf ^= 1u;
  }
}

// Scalar tail for n % 4 != 0 (not hit for the 32x1024x1024 reference shape).
__global__ void bitflip_tail_kernel(const uint32_t* __restrict__ in,
                                    uint32_t* __restrict__ out,
                                    uint32_t start, uint32_t n) {
  const uint32_t i = start + blockIdx.x * blockDim.x + threadIdx.x;
  if (i < n) out[i] = in[i] ^ flip_mask(i);
}

extern "C" void kernel_launch(void* const* d_in, const int* in_sizes, int n_in,
                              void* d_out, int out_size, void* d_ws, size_t ws_size,
                              hipStream_t stream) {
  (void)in_sizes; (void)n_in; (void)d_ws; (void)ws_size;

  const uint32_t n  = (uint32_t)out_size;      // 33,554,432 for the reference
  const uint32_t n4 = n / 4u;
  const uint32_t* in32 = (const uint32_t*)d_in[0];

  if (n4) {
    // 4096 blocks x 8 waves: for the reference shape each lane runs exactly
    // 8 double-buffered b128 iterations; memory-bound, HBM-saturating.
    uint32_t blocks = (n4 + 255u) / 256u;
    if (blocks > 4096u) blocks = 4096u;
    bitflip_async_kernel<<<dim3(blocks), dim3(256), 0, stream>>>(
        (const v4u*)in32, (v4u*)d_out, n4);
  }
  if (n & 3u) {
    bitflip_tail_kernel<<<1, 64, 0, stream>>>(
        in32, (uint32_t*)d_out, n4 * 4u, n);
  }
}